// NeuralNetwork_71004399337967
// MI455X (gfx1250) — compile-verified
//
#include <hip/hip_runtime.h>

// Problem constants (from the reference)
#define NU_ 100000
#define NI_ 50000
#define E_  2000000
#define D_  64
#define NB_ 2
#define NC_ 5

typedef __attribute__((ext_vector_type(2))) float v2f;
typedef __attribute__((ext_vector_type(8))) float v8f;

// ---------------------------------------------------------------------------
// Zero-fill (workspace is poisoned 0xAA; accumulators must start at 0)
// ---------------------------------------------------------------------------
__global__ __launch_bounds__(256) void zero_kernel(float* __restrict__ p, long n) {
    long i = (long)blockIdx.x * blockDim.x + threadIdx.x;
    if (i < n) p[i] = 0.0f;
}

// ---------------------------------------------------------------------------
// WMMA fp32 GEMM: out[row, out_off + col] = (sum_k A[row,k] * B[k,col]) * scale[row]?
// A: [nrows, 64] row-major, B: [64, 64] row-major.
// One wave computes one 16x16 tile via V_WMMA_F32_16X16X4_F32 (K-loop of 16).
// Block = 128 threads = 4 waves = one 16-row x 64-col strip.
//
// VGPR layouts per CDNA5 ISA:
//   A 16x4 : lanes 0-15 hold M=0..15 K={0,1}; lanes 16-31 hold M=0..15 K={2,3}
//   B 4x16 : lanes 0-15 hold N=0..15 K={0,1}; lanes 16-31 hold N=0..15 K={2,3}
//   C/D    : VGPR r -> lanes 0-15: (M=r, N=lane); lanes 16-31: (M=r+8, N=lane-16)
// ---------------------------------------------------------------------------
__global__ __launch_bounds__(128) void gemm64_wmma(const float* __restrict__ A,
                                                   const float* __restrict__ B,
                                                   const float* __restrict__ scale, // may be null
                                                   float* __restrict__ out,
                                                   int nrows, int out_stride, int out_off) {
    const int wave    = threadIdx.x >> 5;   // 0..3 -> 16-col tile
    const int lane    = threadIdx.x & 31;
    const int rowBase = blockIdx.x * 16;
    const int colBase = wave * 16;
    const int halfSel = lane >> 4;          // 0 or 1 (selects K pair)
    const int l16     = lane & 15;

    if (rowBase >= nrows) return;

    const float* __restrict__ Arow = A + (long)(rowBase + l16) * 64;

    v8f acc = {};
    #pragma unroll
    for (int k = 0; k < 64; k += 4) {
        v2f a, b;
        const int ka = k + 2 * halfSel;
        a.x = Arow[ka + 0];
        a.y = Arow[ka + 1];
        b.x = B[(ka + 0) * 64 + colBase + l16];
        b.y = B[(ka + 1) * 64 + colBase + l16];
        // 8 args: (neg_a, A, neg_b, B, c_mod, C, reuse_a, reuse_b)
        acc = __builtin_amdgcn_wmma_f32_16x16x4_f32(
            false, a, false, b, (short)0, acc, false, false);
    }

    #pragma unroll
    for (int r = 0; r < 8; ++r) {
        const int row = rowBase + r + 8 * halfSel;
        float v = acc[r];
        if (scale) v *= scale[row];   // c has shape [N,1] -> scale[row]
        out[(long)row * out_stride + out_off + colBase + l16] = v;
    }
}

// ---------------------------------------------------------------------------
// Fused bidirectional scatter:
//   h_item[dst[e]] += msg_u[src[e]]   and   h_user[src[e]] += msg_i[dst[e]]
// One thread per (edge, dim). Hardware fp32 atomics; accumulators fit in the
// 192MB L2, so these atomics stay on-chip.
// ---------------------------------------------------------------------------
__global__ __launch_bounds__(256) void scatter_kernel(const float* __restrict__ msg_u,
                                                      const float* __restrict__ msg_i,
                                                      const int* __restrict__ src,
                                                      const int* __restrict__ dst,
                                                      float* __restrict__ h_item,
                                                      float* __restrict__ h_user,
                                                      long total) {
    long i = (long)blockIdx.x * blockDim.x + threadIdx.x;
    if (i >= total) return;
    const int e = (int)(i >> 6);
    const int d = (int)(i & 63);
    const int s = src[e];
    const int t = dst[e];
    unsafeAtomicAdd(&h_item[(long)t * 64 + d], msg_u[(long)s * 64 + d]);
    unsafeAtomicAdd(&h_user[(long)s * 64 + d], msg_i[(long)t * 64 + d]);
}

// h[row, :] *= c[row]
__global__ __launch_bounds__(256) void scale_rows(float* __restrict__ h,
                                                  const float* __restrict__ c, long n) {
    long i = (long)blockIdx.x * blockDim.x + threadIdx.x;
    if (i < n) h[i] *= c[i >> 6];
}

// ---------------------------------------------------------------------------
// Decoder: one wave32 per edge.
//   sr[b] = dot(hu_b[src, b, :], h_item[dst, :])  (64-dim, 2 floats/lane)
//   out[e, c] = sr[0]*Wc[c,0] + sr[1]*Wc[c,1]
// Lane L loads float2 at offset 2L -> fully coalesced 256B row reads.
// ---------------------------------------------------------------------------
__global__ __launch_bounds__(256) void edge_kernel(const float* __restrict__ hu_b,   // [NU,2,64]
                                                   const float* __restrict__ h_item, // [NI,64]
                                                   const int* __restrict__ src,
                                                   const int* __restrict__ dst,
                                                   const float* __restrict__ Wc,     // [5,2]
                                                   float* __restrict__ out,
                                                   int nedges) {
    const int wave = threadIdx.x >> 5;
    const int lane = threadIdx.x & 31;
    const int e = blockIdx.x * 8 + wave;
    if (e >= nedges) return;

    const int s = src[e];
    const int t = dst[e];
    const float* __restrict__ hi = h_item + (long)t * 64;
    const float* __restrict__ hb = hu_b + (long)s * 128;

    const float2 x  = *(const float2*)(hi + lane * 2);
    const float2 a0 = *(const float2*)(hb + lane * 2);
    const float2 a1 = *(const float2*)(hb + 64 + lane * 2);

    float p0 = a0.x * x.x + a0.y * x.y;
    float p1 = a1.x * x.x + a1.y * x.y;

    #pragma unroll
    for (int m = 16; m >= 1; m >>= 1) {
        p0 += __shfl_xor(p0, m, 32);
        p1 += __shfl_xor(p1, m, 32);
    }

    if (lane == 0) {
        #pragma unroll
        for (int c = 0; c < NC_; ++c)
            out[(long)e * NC_ + c] = p0 * Wc[c * 2 + 0] + p1 * Wc[c * 2 + 1];
    }
}

// ---------------------------------------------------------------------------
// Launch
// ---------------------------------------------------------------------------
extern "C" void kernel_launch(void* const* d_in, const int* in_sizes, int n_in,
                              void* d_out, int out_size, void* d_ws, size_t ws_size,
                              hipStream_t stream) {
    const float* ufeat  = (const float*)d_in[0];
    const float* ifeat  = (const float*)d_in[1];
    const float* c_u    = (const float*)d_in[2];
    const float* c_i    = (const float*)d_in[3];
    const float* W_user = (const float*)d_in[4];
    const float* W_item = (const float*)d_in[5];
    const float* P      = (const float*)d_in[6];   // [2,64,64]
    const float* Wc     = (const float*)d_in[7];   // [5,2]
    const int*   src    = (const int*)d_in[8];
    const int*   dst    = (const int*)d_in[9];
    float* out = (float*)d_out;

    // Workspace layout (floats): msg_u | msg_i | h_item | h_user | hu_b
    float* ws     = (float*)d_ws;
    float* msg_u  = ws;                              // NU*64
    float* msg_i  = msg_u + (long)NU_ * 64;          // NI*64
    float* h_item = msg_i + (long)NI_ * 64;          // NI*64
    float* h_user = h_item + (long)NI_ * 64;         // NU*64
    float* hu_b   = h_user + (long)NU_ * 64;         // NU*128

    // 1) zero the accumulators (h_item + h_user are contiguous)
    const long nz = (long)NI_ * 64 + (long)NU_ * 64;
    zero_kernel<<<(int)((nz + 255) / 256), 256, 0, stream>>>(h_item, nz);

    // 2) msg_u = (ufeat @ W_user) * c_u ; msg_i = (ifeat @ W_item) * c_i
    gemm64_wmma<<<NU_ / 16, 128, 0, stream>>>(ufeat, W_user, c_u, msg_u, NU_, 64, 0);
    gemm64_wmma<<<NI_ / 16, 128, 0, stream>>>(ifeat, W_item, c_i, msg_i, NI_, 64, 0);

    // 3) scatter both directions over the 2M edges
    const long total = (long)E_ * 64;
    scatter_kernel<<<(int)((total + 255) / 256), 256, 0, stream>>>(
        msg_u, msg_i, src, dst, h_item, h_user, total);

    // 4) h_item *= c_i ; h_user *= c_u
    scale_rows<<<(int)(((long)NI_ * 64 + 255) / 256), 256, 0, stream>>>(h_item, c_i, (long)NI_ * 64);
    scale_rows<<<(int)(((long)NU_ * 64 + 255) / 256), 256, 0, stream>>>(h_user, c_u, (long)NU_ * 64);

    // 5) hu_b[:, b, :] = h_user @ P[b]  (interleaved output, stride 128)
    gemm64_wmma<<<NU_ / 16, 128, 0, stream>>>(h_user, P,           nullptr, hu_b, NU_, 128, 0);
    gemm64_wmma<<<NU_ / 16, 128, 0, stream>>>(h_user, P + 64 * 64, nullptr, hu_b, NU_, 128, 64);

    // 6) per-edge bilinear decode -> out [E,5]
    edge_kernel<<<(E_ + 7) / 8, 256, 0, stream>>>(hu_b, h_item, src, dst, Wc, out, E_);
}